// LightweightEncoder_62947040690293
// MI455X (gfx1250) — compile-verified
//
#include <hip/hip_runtime.h>

typedef __attribute__((ext_vector_type(2))) float v2f;
typedef __attribute__((ext_vector_type(8))) float v8f;

#define BATCH 16
#define CIN   3
#define HIN   512
#define WIN   512
#define COUT  64
#define HOUT  128
#define WOUT  128
#define KK    48   // CIN*4*4

// ---------------- Kernel 1: exact spectral norm via power iteration ----------------
__global__ __launch_bounds__(64)
void sn_kernel(const float* __restrict__ w, float* __restrict__ wn) {
    __shared__ float W[COUT * KK];
    __shared__ float v[KK];
    __shared__ float u[COUT];
    __shared__ float scale;
    const int tid = threadIdx.x;

    for (int i = tid; i < COUT * KK; i += 64) W[i] = w[i];
    if (tid < KK) v[tid] = 1.0f;
    __syncthreads();

    for (int it = 0; it < 32; ++it) {
        if (tid < COUT) {
            float s = 0.f;
            #pragma unroll 8
            for (int k = 0; k < KK; ++k) s += W[tid * KK + k] * v[k];
            u[tid] = s;
        }
        __syncthreads();
        if (tid < KK) {
            float s = 0.f;
            #pragma unroll 8
            for (int i = 0; i < COUT; ++i) s += W[i * KK + tid] * u[i];
            v[tid] = s;
        }
        __syncthreads();
        if (tid == 0) {
            float n = 0.f;
            for (int k = 0; k < KK; ++k) n += v[k] * v[k];
            scale = rsqrtf(n);
        }
        __syncthreads();
        if (tid < KK) v[tid] *= scale;
        __syncthreads();
    }
    // sigma = ||W v||, ||v|| == 1
    if (tid < COUT) {
        float s = 0.f;
        #pragma unroll 8
        for (int k = 0; k < KK; ++k) s += W[tid * KK + k] * v[k];
        u[tid] = s;
    }
    __syncthreads();
    if (tid == 0) {
        float n = 0.f;
        for (int i = 0; i < COUT; ++i) n += u[i] * u[i];
        scale = rsqrtf(n);          // 1/sigma
    }
    __syncthreads();
    const float inv_sigma = scale;
    for (int i = tid; i < COUT * KK; i += 64) wn[i] = W[i] * inv_sigma;
}

// ---------------- Kernel 2: fused conv-GEMM (f32 WMMA) + bias + soft-VQ ----------------
// Block = one spatial tile of 16 w'-positions at (b, h'); 4 waves x 16 channels.
// GEMM per wave: D[16ch x 16spatial] = sum_{t=0..11} A(16x4 weights) x B(4x16 patches).
__global__ __launch_bounds__(128)
void conv_vq_kernel(const float* __restrict__ x,
                    const float* __restrict__ wn,
                    const float* __restrict__ bias,
                    const float* __restrict__ centers,
                    float* __restrict__ out) {
    const int lane = threadIdx.x & 31;
    const int wave = threadIdx.x >> 5;

    const int tile = blockIdx.x;                 // 16384 tiles
    const int w0   = (tile & 7) << 4;            // 8 tiles of 16 along W'
    const int hp   = (tile >> 3) & 127;          // H'
    const int b    = tile >> 10;                 // batch

    const int c0   = wave << 4;                  // channel group base
    const int l15  = lane & 15;                  // A: row (channel), B: col (spatial)
    const int koff = (lane < 16) ? 0 : 2;        // K sub-pair per lane half

    // A fragments: weights, constant for the whole wave
    v2f afrag[12];
    const float* wrow = wn + (c0 + l15) * KK + koff;
    #pragma unroll
    for (int t = 0; t < 12; ++t)
        afrag[t] = *(const v2f*)(wrow + 4 * t);

    // B fragments: input patches. k = c*16 + kh*4 + kw ; t = c*4 + kh ; kw = koff..koff+1
    v2f bfrag[12];
    const int colBase = ((w0 + l15) << 2) + koff;   // 4*w' + kw
    #pragma unroll
    for (int t = 0; t < 12; ++t) {
        const int c  = t >> 2;
        const int kh = t & 3;
        const float* p = x + (((b * CIN + c) * HIN) + ((hp << 2) + kh)) * WIN + colBase;
        bfrag[t] = *(const v2f*)p;
    }

    // 12 chained f32 WMMAs: D = A x B + C
    v8f acc = {};
    #pragma unroll
    for (int t = 0; t < 12; ++t)
        acc = __builtin_amdgcn_wmma_f32_16x16x4_f32(
            false, afrag[t], false, bfrag[t], (short)0, acc, false, false);

    // centers in registers
    float cv[8];
    #pragma unroll
    for (int k = 0; k < 8; ++k) cv[k] = centers[k];

    const int half8 = (lane >> 4) << 3;          // row offset 0 or 8 within D layout
    #pragma unroll
    for (int j = 0; j < 8; ++j) {
        const int ch = c0 + j + half8;           // output channel of this D element
        const float y = acc[j] + bias[ch];

        // soft VQ: softmax(-(y-c_k)^2) weighted sum of centers (min-shifted, stable)
        float d[8];
        float dmin = 3.402823466e38f;
        #pragma unroll
        for (int k = 0; k < 8; ++k) {
            const float t0 = y - cv[k];
            d[k] = t0 * t0;
            dmin = fminf(dmin, d[k]);
        }
        float s = 0.f, num = 0.f;
        #pragma unroll
        for (int k = 0; k < 8; ++k) {
            const float e = __expf(dmin - d[k]);
            s   += e;
            num += cv[k] * e;
        }
        // NCHW store: lanes 0-15 / 16-31 each write 16 consecutive w' floats
        out[(((b * COUT + ch) * HOUT) + hp) * WOUT + w0 + l15] = num / s;
    }
}

extern "C" void kernel_launch(void* const* d_in, const int* in_sizes, int n_in,
                              void* d_out, int out_size, void* d_ws, size_t ws_size,
                              hipStream_t stream) {
    (void)in_sizes; (void)n_in; (void)out_size; (void)ws_size;
    const float* x       = (const float*)d_in[0];
    const float* w       = (const float*)d_in[1];
    const float* bias    = (const float*)d_in[2];
    const float* centers = (const float*)d_in[3];
    float* out = (float*)d_out;
    float* wn  = (float*)d_ws;                   // 64*48 floats = 12 KB scratch

    sn_kernel<<<1, 64, 0, stream>>>(w, wn);
    // 16 * 128 * 8 = 16384 spatial tiles; 128 threads = 4 waves (channels 0..63)
    conv_vq_kernel<<<16384, 128, 0, stream>>>(x, wn, bias, centers, out);
}